// TaxaNet_custom_loss_3246995276315
// MI455X (gfx1250) — compile-verified
//
#include <hip/hip_runtime.h>
#include <stdint.h>

#define N_TOT   12020      // 20 + 300 + 2700 + 9000
#define NF4_ROW 3005       // 12020 / 4
#define NF4_TOT 6010       // two contiguous rows (l=2, l=3)
#define BATCH   4096
#define L_DIM   4
#define CHUNK   512        // float4s per chunk (8 KB)
#define NCHUNK  12         // ceil(6010/512)
#define E_CONST 2.718281828459045f

// ---- CDNA5 async-to-LDS path (ASYNCcnt) -------------------------------------
__device__ __forceinline__ uint32_t lds_off_of(const void* p) {
    // AS(3) offsets are the low 32 bits of the generic LDS-aperture address.
    return (uint32_t)(uintptr_t)p;
}
__device__ __forceinline__ void async_ld_b128(uint32_t lds_off, const float4* g) {
    asm volatile("global_load_async_to_lds_b128 %0, %1, off"
                 :: "v"(lds_off), "v"(g) : "memory");
}
__device__ __forceinline__ void wait_async_le2() {
    asm volatile("s_wait_asynccnt 0x2" ::: "memory");
}
__device__ __forceinline__ void wait_async_0() {
    asm volatile("s_wait_asynccnt 0x0" ::: "memory");
}
__device__ __forceinline__ void wait_ds_0() {
    // drain our own pending ds_loads before the async unit may overwrite
    // the buffer they read from (hardware-level WAR on LDS)
    asm volatile("s_wait_dscnt 0x0" ::: "memory");
}

// ---- Kernel 1: per-batch-element row scan -----------------------------------
__global__ __launch_bounds__(256) void taxa_row_kernel(
    const float* __restrict__ y_pred, const float* __restrict__ H,
    const int* __restrict__ y_true,
    float* __restrict__ ws_nll, float* __restrict__ ws_hit)
{
    __shared__ float4 shBuf[2][CHUNK];
    __shared__ float4 shDummy;               // sink for clamped tail lanes
    __shared__ float  red_m2[256]; __shared__ int red_i2[256];
    __shared__ float  red_m3[256]; __shared__ int red_i3[256];
    __shared__ float  red_s3[256];

    const int b   = blockIdx.x;
    const int tid = threadIdx.x;
    // rows l=2 and l=3 are contiguous: start at element (b*4 + 2)*N
    const float4* rowBase =
        (const float4*)(y_pred + ((size_t)b * L_DIM + 2) * N_TOT);

    const uint32_t buf0  = lds_off_of(&shBuf[0][0]);
    const uint32_t buf1  = lds_off_of(&shBuf[1][0]);
    const uint32_t dummy = lds_off_of(&shDummy);

    // issue one chunk: every wave issues exactly 2 async b128 instructions
    auto issue = [&](int c, int sel) {
        wait_ds_0();                          // LDS WAR guard (see above)
        const uint32_t base = sel ? buf1 : buf0;
        int g0 = c * CHUNK + tid;
        int g1 = g0 + 256;
        int cg0 = (g0 < NF4_TOT) ? g0 : (NF4_TOT - 1);
        int cg1 = (g1 < NF4_TOT) ? g1 : (NF4_TOT - 1);
        uint32_t l0 = (g0 < NF4_TOT) ? (base + (uint32_t)tid * 16u)          : dummy;
        uint32_t l1 = (g1 < NF4_TOT) ? (base + ((uint32_t)tid + 256u) * 16u) : dummy;
        async_ld_b128(l0, rowBase + cg0);
        async_ld_b128(l1, rowBase + cg1);
    };

    float m2 = -INFINITY, m3 = -INFINITY, s3 = 0.0f;
    int   i2 = 0, i3 = 0;

    auto consume = [&](int g, const float4* slot) {
        if (g >= NF4_TOT) return;
        const float4 v = *slot;
        const float x0 = v.x, x1 = v.y, x2 = v.z, x3 = v.w;
        const float mx = fmaxf(fmaxf(x0, x1), fmaxf(x2, x3));
        const int p = g * 4;
        if (g < NF4_ROW) {                       // row l=2: plain argmax
            if (mx > m2) {
                m2 = mx;
                i2 = p + ((x0 == mx) ? 0 : (x1 == mx) ? 1 : (x2 == mx) ? 2 : 3);
            }
        } else {                                 // row l=3: argmax + online logsumexp
            const float mnew = fmaxf(m3, mx);
            const float acc  = __expf(x0 - mnew) + __expf(x1 - mnew)
                             + __expf(x2 - mnew) + __expf(x3 - mnew);
            s3 = s3 * __expf(m3 - mnew) + acc;   // exp(-inf)=0 handles first quad
            if (mx > m3) {
                const int q = p - N_TOT;
                i3 = q + ((x0 == mx) ? 0 : (x1 == mx) ? 1 : (x2 == mx) ? 2 : 3);
            }
            m3 = mnew;
        }
    };

    // double-buffered async pipeline; per-wave ASYNCcnt bookkeeping, no barriers
    issue(0, 0);
    for (int c = 0; c < NCHUNK; ++c) {
        const int sel = c & 1;
        if (c + 1 < NCHUNK) { issue(c + 1, sel ^ 1); wait_async_le2(); }
        else                { wait_async_0(); }
        consume(c * CHUNK + tid,       &shBuf[sel][tid]);
        consume(c * CHUNK + 256 + tid, &shBuf[sel][tid + 256]);
    }

    // deterministic block tree reduction
    red_m2[tid] = m2; red_i2[tid] = i2;
    red_m3[tid] = m3; red_i3[tid] = i3; red_s3[tid] = s3;
    __syncthreads();
    for (int st = 128; st > 0; st >>= 1) {
        if (tid < st) {
            float ma = red_m2[tid], mb = red_m2[tid + st];
            int   ia = red_i2[tid], ib = red_i2[tid + st];
            if (mb > ma || (mb == ma && ib < ia)) { red_m2[tid] = mb; red_i2[tid] = ib; }

            float na = red_m3[tid], nb = red_m3[tid + st];
            int   ja = red_i3[tid], jb = red_i3[tid + st];
            float sa = red_s3[tid], sb = red_s3[tid + st];
            float mm = fmaxf(na, nb);
            red_s3[tid] = sa * __expf(na - mm) + sb * __expf(nb - mm);
            red_m3[tid] = mm;
            red_i3[tid] = (nb > na || (nb == na && jb < ja)) ? jb : ja;
        }
        __syncthreads();
    }

    if (tid == 0) {
        const int idx2 = red_i2[0];
        const int idx3 = red_i3[0];
        const int yt   = y_true[b * L_DIM + 3];
        const float tv = y_pred[((size_t)b * L_DIM + 3) * N_TOT + yt];
        const float nll = red_m3[0] + logf(red_s3[0]) - tv;
        const float hv  = H[(size_t)idx3 * N_TOT + idx2];
        ws_nll[b] = nll;
        ws_hit[b] = (hv == 1.0f) ? 1.0f : 0.0f;
    }
}

// ---- Kernel 2: deterministic final reduction --------------------------------
__global__ __launch_bounds__(256) void taxa_final_kernel(
    const float* __restrict__ ws_nll, const float* __restrict__ ws_hit,
    float* __restrict__ out)
{
    __shared__ float sn[256], sh[256];
    const int tid = threadIdx.x;
    float a = 0.0f, h = 0.0f;
    for (int i = tid; i < BATCH; i += 256) { a += ws_nll[i]; h += ws_hit[i]; }
    sn[tid] = a; sh[tid] = h;
    __syncthreads();
    for (int st = 128; st > 0; st >>= 1) {
        if (tid < st) { sn[tid] += sn[tid + st]; sh[tid] += sh[tid + st]; }
        __syncthreads();
    }
    if (tid == 0) {
        const float ce   = sn[0] / (float)BATCH;
        const float hier = sh[0] * E_CONST;
        out[0] = (hier + ce) * 0.1f;
    }
}

extern "C" void kernel_launch(void* const* d_in, const int* in_sizes, int n_in,
                              void* d_out, int out_size, void* d_ws, size_t ws_size,
                              hipStream_t stream) {
    (void)in_sizes; (void)n_in; (void)out_size; (void)ws_size;
    const float* y_pred = (const float*)d_in[0];
    const float* H      = (const float*)d_in[1];
    const int*   y_true = (const int*)d_in[2];
    float* ws_nll = (float*)d_ws;
    float* ws_hit = ws_nll + BATCH;

    taxa_row_kernel<<<BATCH, 256, 0, stream>>>(y_pred, H, y_true, ws_nll, ws_hit);
    taxa_final_kernel<<<1, 256, 0, stream>>>(ws_nll, ws_hit, (float*)d_out);
}